// GaussianNeighborAssociator_81346680586528
// MI455X (gfx1250) — compile-verified
//
#include <hip/hip_runtime.h>

// ---------------------------------------------------------------------------
// GaussianNeighborAssociator for MI455X (gfx1250, wave32)
//
// d2 = A(Mx8) * B(8xN) as a K=7 GEMM via V_WMMA_F32_16X16X4_F32 (two chained
// K=4 WMMAs per 16x16 tile), fused with a deterministic radix-select top-64:
//   phase 1: WMMA d2 tiles -> per-row 512-bin LDS histogram (branchless,
//            trash bin NBINS-1 for masked-out elements)
//   phase 2: per-row scan -> threshold bin + below-count
//   phase 3: ordered rescan with ballot/prefix-popc slot assignment
//
// A and B are stored fragment-permuted [k0,k1,k4,k5, k2,k3,k6,k7] so each
// lane's two WMMA K-fragments come from ONE global_load_b128.
// ---------------------------------------------------------------------------

typedef __attribute__((ext_vector_type(2))) float v2f;
typedef __attribute__((ext_vector_type(4))) float v4f;
typedef __attribute__((ext_vector_type(8))) float v8f;

#define KSEL 64
#define NBINS 512
#define ROWS_PER_WG 16
#define NWAVES 4
#define TPB (NWAVES * 32)

// Permuted storage order: p[0..7] = k0,k1,k4,k5,k2,k3,k6,k7
// Inverse map (logical k -> physical p): k0->0 k1->1 k2->4 k3->5 k4->2 k5->3 k6->6

// ---------------- Gaussian preprocessing: permuted B + radii ----------------
__global__ void prep_gaussians_kernel(const float* __restrict__ mu,
                                      const float* __restrict__ scale,
                                      float* __restrict__ Bm,
                                      float* __restrict__ radii,
                                      int N) {
  int i = blockIdx.x * blockDim.x + threadIdx.x;
  if (i >= N) return;
  const float eps = 1e-8f;
  float sx = scale[i * 3 + 0], sy = scale[i * 3 + 1], sz = scale[i * 3 + 2];
  float sx2 = sx * sx, sy2 = sy * sy, sz2 = sz * sz;
  float ivx = 1.0f / (sx2 + eps), ivy = 1.0f / (sy2 + eps), ivz = 1.0f / (sz2 + eps);
  float mx = mu[i * 3 + 0], my = mu[i * 3 + 1], mz = mu[i * 3 + 2];
  float mix = mx * ivx, miy = my * ivy, miz = mz * ivz;
  float mterm = mx * mix + my * miy + mz * miz;
  radii[i] = 9.0f * (sx2 + sy2 + sz2);  // SCALE_MULT^2 * sum(scale^2)
  // logical k: [ivx, ivy, ivz, -2mix, -2miy, -2miz, mterm, 0]
  v4f lo = {ivx, ivy, -2.0f * miy, -2.0f * miz};   // k0,k1,k4,k5
  v4f hi = {ivz, -2.0f * mix, mterm, 0.0f};        // k2,k3,k6,k7
  *(v4f*)(Bm + (size_t)i * 8)     = lo;
  *(v4f*)(Bm + (size_t)i * 8 + 4) = hi;
}

// ---------------- Voxel preprocessing: centers output + permuted A ----------
__global__ void centers_A_kernel(const int* __restrict__ voxel,
                                 const float* __restrict__ pcr,
                                 float* __restrict__ centers_out,
                                 float* __restrict__ Am,
                                 int M) {
  int i = blockIdx.x * blockDim.x + threadIdx.x;
  if (i >= M) return;
  float xi = (float)voxel[i * 4 + 3];
  float yi = (float)voxel[i * 4 + 2];
  float zi = (float)voxel[i * 4 + 1];
  float cx = pcr[0] + (xi + 0.5f) * 0.2f;
  float cy = pcr[1] + (yi + 0.5f) * 0.2f;
  float cz = pcr[2] + (zi + 0.5f) * 0.2f;
  centers_out[i * 3 + 0] = cx;
  centers_out[i * 3 + 1] = cy;
  centers_out[i * 3 + 2] = cz;
  // logical k: [cx2, cy2, cz2, cx, cy, cz, 1, 0]
  v4f lo = {cx * cx, cy * cy, cy, cz};   // k0,k1,k4,k5
  v4f hi = {cz * cz, cx, 1.0f, 0.0f};    // k2,k3,k6,k7
  *(v4f*)(Am + (size_t)i * 8)     = lo;
  *(v4f*)(Am + (size_t)i * 8 + 4) = hi;
}

// ---------------- helpers ---------------------------------------------------
__device__ __forceinline__ void write_slot(void* idx_out, float* w_out, float* m_out,
                                           int idx_wide, long long row, int slot,
                                           long long idx, float w, float mk) {
  long long o = row * KSEL + slot;
  if (idx_wide) ((long long*)idx_out)[o] = idx;
  else          ((int*)idx_out)[o] = (int)idx;
  w_out[o] = w;
  m_out[o] = mk;
}

// ---------------- main fused kernel: WMMA distances + radix-select top-64 ---
__launch_bounds__(TPB)
__global__ void assoc_topk_kernel(const float* __restrict__ Am,
                                  const float* __restrict__ Bm,
                                  const float* __restrict__ radii,
                                  void* __restrict__ idx_out,
                                  float* __restrict__ w_out,
                                  float* __restrict__ m_out,
                                  int idx_wide, int N) {
  __shared__ unsigned int hist[ROWS_PER_WG][NBINS];   // 32 KB
  __shared__ unsigned int thrBin[ROWS_PER_WG];
  __shared__ unsigned int cntBelow[ROWS_PER_WG];

  const int tid  = threadIdx.x;
  const int wave = tid >> 5;
  const int lane = tid & 31;
  const int half = lane >> 4;      // fragment half: K {2h,2h+1} + {2h+4,2h+5}
  const int l16  = lane & 15;
  const long long rowBase = (long long)blockIdx.x * ROWS_PER_WG;

  for (int i = tid; i < ROWS_PER_WG * NBINS; i += TPB)
    ((unsigned int*)hist)[i] = 0u;
  __syncthreads();

  // ---- Phase 1: WMMA distance tiles -> per-row histograms (branchless) ----
  {
    // one b128 per lane supplies both A K-fragments (permuted layout)
    v4f av = *(const v4f*)(Am + (size_t)(rowBase + l16) * 8 + half * 4);
    v2f a_lo = {av.x, av.y};   // K {2h, 2h+1}
    v2f a_hi = {av.z, av.w};   // K {2h+4, 2h+5}

    const int colsPerWave = N / NWAVES;
    const int tiles = colsPerWave >> 4;
    const int cbase = wave * colsPerWave;

    for (int t = 0; t < tiles; ++t) {
      const int col = cbase + t * 16 + l16;
      const float* brow = Bm + (size_t)col * 8 + half * 4;
      v4f bv = *(const v4f*)(brow);
      if (t + 1 < tiles) __builtin_prefetch(brow + 128, 0, 1);  // next B tile
      float rs = radii[col];
      v2f b_lo = {bv.x, bv.y};
      v2f b_hi = {bv.z, bv.w};

      v8f c = {};
      c = __builtin_amdgcn_wmma_f32_16x16x4_f32(false, a_lo, false, b_lo,
                                                (short)0, c, false, false);
      c = __builtin_amdgcn_wmma_f32_16x16x4_f32(false, a_hi, false, b_hi,
                                                (short)0, c, false, false);

      // C layout: lane holds col = c0+l16, rows v + 8*half for VGPR v.
      const unsigned int rowSel = half * 8;
      #pragma unroll
      for (int v = 0; v < 8; ++v) {
        float d2 = c[v];
        unsigned int b = __float_as_uint(fmaxf(d2, 0.0f)) >> 22;
        b = (b > NBINS - 2u) ? (NBINS - 2u) : b;
        unsigned int bin = (d2 <= rs) ? b : (NBINS - 1u);   // trash bin if masked
        atomicAdd(&hist[rowSel + v][bin], 1u);              // unconditional
      }
    }
  }
  __syncthreads();

  // ---- Phase 2: per-row threshold bin (trash bin NBINS-1 excluded) ----
  if (tid < ROWS_PER_WG) {
    unsigned int cum = 0;
    unsigned int T = NBINS - 1;   // default: take every valid element
    for (int b = 0; b < NBINS - 1; ++b) {
      unsigned int h = hist[tid][b];
      if (cum + h >= KSEL) { T = (unsigned int)b; break; }
      cum += h;
    }
    thrBin[tid] = T;
    cntBelow[tid] = cum;          // count strictly below bin T
  }
  __syncthreads();

  // ---- init all output slots for this workgroup ----
  for (int i = tid; i < ROWS_PER_WG * KSEL; i += TPB) {
    long long row = rowBase + (i / KSEL);
    write_slot(idx_out, w_out, m_out, idx_wide, row, i % KSEL, -1, 0.0f, 0.0f);
  }
  __syncthreads();

  // ---- Phase 3: ordered collection, wave-per-4-rows, deterministic ----
  const unsigned int ltm = (1u << lane) - 1u;
  for (int rr = 0; rr < ROWS_PER_WG / NWAVES; ++rr) {
    const int rl = wave * (ROWS_PER_WG / NWAVES) + rr;
    const long long row = rowBase + rl;
    const float* ar = Am + (size_t)row * 8;   // uniform address -> broadcast
    v4f ap0 = *(const v4f*)(ar);
    v4f ap1 = *(const v4f*)(ar + 4);
    // un-permute: k0,k1,k4,k5 in ap0; k2,k3,k6 in ap1.xyz
    float a0 = ap0.x, a1 = ap0.y, a4 = ap0.z, a5 = ap0.w;
    float a2 = ap1.x, a3 = ap1.y, a6 = ap1.z;

    const unsigned int T = thrBin[rl];
    unsigned int belowTot = cntBelow[rl];
    if (belowTot > KSEL) belowTot = KSEL;     // safety (recompute skew)
    unsigned int belowCnt = 0, tieCnt = 0;

    const int batches = N >> 5;
    for (int b = 0; b < batches; ++b) {
      const int col = (b << 5) + lane;
      const float* br = Bm + (size_t)col * 8;
      v4f bp0 = *(const v4f*)(br);       // k0,k1,k4,k5
      v4f bp1 = *(const v4f*)(br + 4);   // k2,k3,k6,k7
      float d2 = a0 * bp0.x;
      d2 = fmaf(a1, bp0.y, d2);
      d2 = fmaf(a2, bp1.x, d2);
      d2 = fmaf(a3, bp1.y, d2);
      d2 = fmaf(a4, bp0.z, d2);
      d2 = fmaf(a5, bp0.w, d2);
      d2 = fmaf(a6, bp1.z, d2);
      bool valid = (d2 <= radii[col]);
      unsigned int bin = __float_as_uint(fmaxf(d2, 0.0f)) >> 22;
      bin = (bin > NBINS - 2u) ? (NBINS - 2u) : bin;
      bool isBelow = valid && (bin < T);
      bool isTie   = valid && (bin == T);

      unsigned int mB = __builtin_amdgcn_ballot_w32(isBelow);
      unsigned int mT = __builtin_amdgcn_ballot_w32(isTie);

      if (isBelow) {
        unsigned int slot = belowCnt + (unsigned int)__popc(mB & ltm);
        if (slot < belowTot)
          write_slot(idx_out, w_out, m_out, idx_wide, row, (int)slot, col, d2, 1.0f);
      } else if (isTie) {
        unsigned int slot = belowTot + tieCnt + (unsigned int)__popc(mT & ltm);
        if (slot < KSEL)
          write_slot(idx_out, w_out, m_out, idx_wide, row, (int)slot, col, d2, 1.0f);
      }
      belowCnt += (unsigned int)__popc(mB);
      tieCnt   += (unsigned int)__popc(mT);
      if (belowCnt >= belowTot && belowTot + tieCnt >= KSEL) break;  // uniform
    }
  }
}

// ---------------------------------------------------------------------------
extern "C" void kernel_launch(void* const* d_in, const int* in_sizes, int n_in,
                              void* d_out, int out_size, void* d_ws, size_t ws_size,
                              hipStream_t stream) {
  const int*   voxel = (const int*)d_in[0];     // (M,4) int32
  const float* mu    = (const float*)d_in[1];   // (N,3) f32
  const float* scale = (const float*)d_in[2];   // (N,3) f32
  const float* pcr   = (const float*)d_in[3];   // (6,)  f32

  const int M = in_sizes[0] / 4;
  const int N = in_sizes[1] / 3;

  // workspace: A (M*8), B (N*8), radii (N)
  float* Am    = (float*)d_ws;
  float* Bm    = Am + (size_t)M * 8;
  float* radii = Bm + (size_t)N * 8;

  // output layout: centers | indices (int64 or int32 slots) | weights | masks
  float* out = (float*)d_out;
  long long wideNeed = (long long)M * 3 + (long long)M * KSEL * 2 +
                       (long long)M * KSEL + (long long)M * KSEL;
  int idx_wide = ((long long)out_size >= wideNeed) ? 1 : 0;
  float* centers_out = out;
  void*  idx_out     = (void*)(out + (size_t)M * 3);
  float* w_out       = out + (size_t)M * 3 + (size_t)M * KSEL * (idx_wide ? 2 : 1);
  float* m_out       = w_out + (size_t)M * KSEL;

  prep_gaussians_kernel<<<(N + 255) / 256, 256, 0, stream>>>(mu, scale, Bm, radii, N);
  centers_A_kernel<<<(M + 255) / 256, 256, 0, stream>>>(voxel, pcr, centers_out, Am, M);
  assoc_topk_kernel<<<M / ROWS_PER_WG, TPB, 0, stream>>>(Am, Bm, radii, idx_out,
                                                         w_out, m_out, idx_wide, N);
}